// DPS_topk_86638080295020
// MI455X (gfx1250) — compile-verified
//
#include <hip/hip_runtime.h>

// Problem constants (from the reference)
#define SAMPLES 32
#define ROWS    64
#define DIM     1024
#define KTOP    16
#define TEMPR   5.0f
// exp(log(1e-20)/TEMP) == 1e-4 : multiplicative suppression of masked positions
#define ALPHA   1.0e-4f

typedef float v2f __attribute__((ext_vector_type(2)));
typedef float v8f __attribute__((ext_vector_type(8)));

// Order-preserving float -> u32 mapping (for packed u64 argmax keys)
__device__ __forceinline__ unsigned int f32_mono(float f) {
    unsigned int b = __float_as_uint(f);
    return b ^ ((b & 0x80000000u) ? 0xFFFFFFFFu : 0x80000000u);
}
__device__ __forceinline__ float mono_f32(unsigned int m) {
    unsigned int b = (m & 0x80000000u) ? (m ^ 0x80000000u) : ~m;
    return __uint_as_float(b);
}

__global__ __launch_bounds__(256)
void dps_topk_fused(const float* __restrict__ logits,
                    const float* __restrict__ gn,
                    float* __restrict__ out) {
    const int bx  = blockIdx.x;          // 0..2047  == s*ROWS + r
    const int r   = bx & (ROWS - 1);
    const int tid = threadIdx.x;         // 256 threads, 4 elements each
    const int d0  = tid * 4;

    __shared__ unsigned long long s_key;     // packed (value<<32 | idx) argmax cell
    __shared__ int   s_idx[KTOP];            // selected indices (later: ascending)
    __shared__ float s_val[KTOP];            // their perturbed values
    __shared__ float s_wt[KTOP];             // exp((val - max)/T) in index order
    __shared__ float s_P[KTOP];              // prefix sums P[k] = sum_{j<k} wt[j]
    __shared__ float s_b;                    // row max (softmax shift)
    __shared__ float s_red[256];             // block sum reduction

    const float* gn_row = gn     + (size_t)bx * DIM;
    const float* lg_row = logits + (size_t)r  * DIM;

    // gfx1250 prefetch path (global_prefetch_b8)
    __builtin_prefetch(lg_row + d0, 0, 0);

    const float4 g4 = *(const float4*)(gn_row + d0);
    const float4 l4 = *(const float4*)(lg_row + d0);
    float p[4] = { l4.x + g4.x, l4.y + g4.y, l4.z + g4.z, l4.w + g4.w };

    // ---- Top-16 extraction: 16 rounds of block argmax via LDS u64 atomicMax ----
    int chosen = 0;  // per-thread mask over its 4 elements
    for (int it = 0; it < KTOP; ++it) {
        if (tid == 0) s_key = 0ull;
        __syncthreads();
        unsigned long long best = 0ull;
        #pragma unroll
        for (int j = 0; j < 4; ++j) {
            if (!(chosen & (1 << j))) {
                unsigned long long key =
                    ((unsigned long long)f32_mono(p[j]) << 32) | (unsigned int)(d0 + j);
                if (key > best) best = key;
            }
        }
        if (best) atomicMax(&s_key, best);       // ds_max_u64 on LDS
        __syncthreads();
        const unsigned long long win = s_key;
        const int widx = (int)(win & 0xFFFFFFFFu);
        if (tid == 0) { s_idx[it] = widx; s_val[it] = mono_f32((unsigned int)(win >> 32)); }
        if ((widx >> 2) == tid) chosen |= 1 << (widx & 3);
        __syncthreads();
    }

    // ---- Sort selected indices ascending (reference: top_k(idx)[..., ::-1]) ----
    if (tid == 0) {
        s_b = s_val[0];                          // extraction order => [0] is the row max
        for (int i = 1; i < KTOP; ++i) {
            int ii = s_idx[i]; float vv = s_val[i];
            int j = i - 1;
            while (j >= 0 && s_idx[j] > ii) { s_idx[j+1] = s_idx[j]; s_val[j+1] = s_val[j]; --j; }
            s_idx[j+1] = ii; s_val[j+1] = vv;
        }
        const float b = s_b;
        for (int i = 0; i < KTOP; ++i)
            s_wt[i] = __expf((s_val[i] - b) * (1.0f / TEMPR));
    }
    __syncthreads();

    // ---- P[k] = sum_{j<k} wt[j] as L(16x16, strictly lower) x wt via WMMA f32 ----
    // 4 chained V_WMMA_F32_16X16X4_F32, K=4 per step, exact in f32.
    // A 16x4 layout: lane l -> M = l&15 ; lanes 0-15 hold K={0,1}, lanes 16-31 K={2,3}.
    // B 4x16: row K broadcast across N (wt[k] in every column), same K striping.
    // D 16x16: acc[v] on lane l = D[v + 8*(l>>4)][l&15]; column n==0 holds P[m].
    if (tid < 32) {
        const int lane = tid;
        const int m    = lane & 15;
        const int kh   = lane >> 4;
        v8f acc = {};
        #pragma unroll
        for (int c = 0; c < 4; ++c) {
            const int k0 = 4 * c + 2 * kh;
            v2f a, b;
            a[0] = (k0     < m) ? 1.0f : 0.0f;   // L[m][k0]
            a[1] = (k0 + 1 < m) ? 1.0f : 0.0f;   // L[m][k0+1]
            b[0] = s_wt[k0];
            b[1] = s_wt[k0 + 1];
            acc = __builtin_amdgcn_wmma_f32_16x16x4_f32(
                      false, a, false, b, (short)0, acc, false, false);
        }
        if ((lane & 15) == 0) {                  // lanes 0 and 16 hold column n=0
            #pragma unroll
            for (int v = 0; v < 8; ++v)
                s_P[v + 8 * kh] = acc[v];
        }
    }
    __syncthreads();

    // ---- Row softmax weights + block sum S ----
    const float b = s_b;
    float w[4], tsum = 0.0f;
    #pragma unroll
    for (int j = 0; j < 4; ++j) {
        w[j] = __expf((p[j] - b) * (1.0f / TEMPR));
        tsum += w[j];
    }
    s_red[tid] = tsum;
    __syncthreads();
    for (int off = 128; off > 0; off >>= 1) {
        if (tid < off) s_red[tid] += s_red[tid + off];
        __syncthreads();
    }
    const float S = s_red[0];

    // rank[j]: slot of element d0+j among the ascending-sorted selections, 16 if absent
    int rank[4];
    #pragma unroll
    for (int j = 0; j < 4; ++j) {
        rank[j] = KTOP;
        #pragma unroll
        for (int t = 0; t < KTOP; ++t)
            if (s_idx[t] == d0 + j) rank[j] = t;
    }

    // ---- Stream the (K, D) output tile: pure store-bandwidth phase ----
    float* out_row = out + (size_t)bx * (KTOP * DIM);
    for (int k = 0; k < KTOP; ++k) {
        const float inv = 1.0f / (S - (1.0f - ALPHA) * s_P[k]);
        float4 o;
        float* op = (float*)&o;
        #pragma unroll
        for (int j = 0; j < 4; ++j) {
            const float soft = w[j] * ((rank[j] < k) ? ALPHA : 1.0f) * inv;
            const float hard = (rank[j] == k) ? 1.0f : 0.0f;
            op[j] = (hard - soft) + soft;        // forward value of stop_gradient trick
        }
        *(float4*)(out_row + (size_t)k * DIM + d0) = o;
    }
}

extern "C" void kernel_launch(void* const* d_in, const int* in_sizes, int n_in,
                              void* d_out, int out_size, void* d_ws, size_t ws_size,
                              hipStream_t stream) {
    const float* logits = (const float*)d_in[0];   // (64, 1024) f32
    const float* gn     = (const float*)d_in[1];   // (32, 64, 1024) f32
    float* out          = (float*)d_out;           // (32, 64, 16, 1024) f32
    (void)in_sizes; (void)n_in; (void)out_size; (void)d_ws; (void)ws_size;
    dps_topk_fused<<<dim3(SAMPLES * ROWS), dim3(256), 0, stream>>>(logits, gn, out);
}